// NTXentLossChildrenFmapsClean_85624468013610
// MI455X (gfx1250) — compile-verified
//
#include <hip/hip_runtime.h>

#define F_ 3
#define B_ 64
#define C_ 16
#define D_ 512
#define ROWS (B_*C_)          // 1024 rows per fmap bank per f
#define TEMP_INV 10.0f        // 1/TEMPERATURE
#define EPSV 1e-8f
#define NEG_INF_V -1.0e30f

typedef __attribute__((ext_vector_type(2))) float v2f;
typedef __attribute__((ext_vector_type(8))) float v8f;

// ---------------------------------------------------------------------------
// Kernel 1: per-row squared norms  Ni[f,b,c] = sum_d Xi^2 , Nj likewise.
// One wave32 per row of 512 floats (16 coalesced loads/lane + shfl reduce).
// ---------------------------------------------------------------------------
__global__ void row_norms_kernel(const float* __restrict__ Xi,
                                 const float* __restrict__ Xj,
                                 float* __restrict__ Ni,
                                 float* __restrict__ Nj) {
    int wave = (blockIdx.x * blockDim.x + threadIdx.x) >> 5;
    int lane = threadIdx.x & 31;
    const int total = 2 * F_ * ROWS;          // 6144 rows
    if (wave >= total) return;
    const float* src; float* dst; int r;
    if (wave < F_ * ROWS) { src = Xi; dst = Ni; r = wave; }
    else                  { src = Xj; dst = Nj; r = wave - F_ * ROWS; }
    const float* row = src + (size_t)r * D_;
    float s = 0.f;
    #pragma unroll
    for (int t = 0; t < D_ / 32; ++t) {
        float x = row[lane + 32 * t];
        s += x * x;
    }
    #pragma unroll
    for (int m = 16; m >= 1; m >>= 1) s += __shfl_xor(s, m, 32);
    if (lane == 0) dst[r] = s;
}

// ---------------------------------------------------------------------------
// Kernel 2: pairwise row-dot GEMMs via V_WMMA_F32_16X16X4_F32.
//   which==0 : P[f] = Xi[f] (1024x512)  x  Xj[f]^T  -> 1024x1024
//   which==1 : Q[f] = Xi[f]            x  Xi[f]^T  -> 1024x1024
// Register-blocked: one wave owns a 32x64 output strip = 2x4 WMMA tiles.
// Per K-step of 4: 6 float2 loads feed 8 WMMAs (A frags reused 4x, B 2x).
// A-frag lane layout (32-bit 16x4): m = lane&15, K pair = 2*(lane>>4).
// global_prefetch_b8 issued one 64-float chunk ahead per row pointer.
// No early-exit: EXEC all-1s as WMMA requires.
// ---------------------------------------------------------------------------
__global__ void gemm_wmma_kernel(const float* __restrict__ Xi,
                                 const float* __restrict__ Xj,
                                 float* __restrict__ P,
                                 float* __restrict__ Q) {
    int gw   = (blockIdx.x * blockDim.x + threadIdx.x) >> 5;
    int lane = threadIdx.x & 31;
    int tng   = gw & 15; gw >>= 4;   // N group: 4 tiles -> 64 cols
    int tmg   = gw & 31; gw >>= 5;   // M group: 2 tiles -> 32 rows
    int which = gw & 1;  gw >>= 1;
    int f     = gw;                  // 0..2

    const float* abase = Xi + (size_t)f * ROWS * D_;
    const float* bbase = ((which == 0) ? Xj : Xi) + (size_t)f * ROWS * D_;
    const int lm = lane & 15;
    const int kh = (lane >> 4) * 2;  // K sub-pair selector

    const float* A0 = abase + (size_t)(tmg * 32 + lm) * D_ + kh;
    const float* A1 = A0 + (size_t)16 * D_;
    const float* B0 = bbase + (size_t)(tng * 64 + lm) * D_ + kh;
    const float* B1 = B0 + (size_t)16 * D_;
    const float* B2 = B0 + (size_t)32 * D_;
    const float* B3 = B0 + (size_t)48 * D_;

    v8f acc[2][4] = {};
    for (int kc = 0; kc < D_; kc += 16) {
        if (kc + 64 < D_) {            // uniform branch; speculative prefetch
            __builtin_prefetch(A0 + kc + 64, 0, 1);
            __builtin_prefetch(A1 + kc + 64, 0, 1);
            __builtin_prefetch(B0 + kc + 64, 0, 1);
            __builtin_prefetch(B1 + kc + 64, 0, 1);
            __builtin_prefetch(B2 + kc + 64, 0, 1);
            __builtin_prefetch(B3 + kc + 64, 0, 1);
        }
        #pragma unroll
        for (int ks = 0; ks < 16; ks += 4) {
            const int k = kc + ks;
            v2f a0 = *(const v2f*)(A0 + k);
            v2f a1 = *(const v2f*)(A1 + k);
            v2f b0 = *(const v2f*)(B0 + k);
            v2f b1 = *(const v2f*)(B1 + k);
            v2f b2 = *(const v2f*)(B2 + k);
            v2f b3 = *(const v2f*)(B3 + k);
            acc[0][0] = __builtin_amdgcn_wmma_f32_16x16x4_f32(false, a0, false, b0, (short)0, acc[0][0], false, false);
            acc[0][1] = __builtin_amdgcn_wmma_f32_16x16x4_f32(false, a0, false, b1, (short)0, acc[0][1], false, false);
            acc[0][2] = __builtin_amdgcn_wmma_f32_16x16x4_f32(false, a0, false, b2, (short)0, acc[0][2], false, false);
            acc[0][3] = __builtin_amdgcn_wmma_f32_16x16x4_f32(false, a0, false, b3, (short)0, acc[0][3], false, false);
            acc[1][0] = __builtin_amdgcn_wmma_f32_16x16x4_f32(false, a1, false, b0, (short)0, acc[1][0], false, false);
            acc[1][1] = __builtin_amdgcn_wmma_f32_16x16x4_f32(false, a1, false, b1, (short)0, acc[1][1], false, false);
            acc[1][2] = __builtin_amdgcn_wmma_f32_16x16x4_f32(false, a1, false, b2, (short)0, acc[1][2], false, false);
            acc[1][3] = __builtin_amdgcn_wmma_f32_16x16x4_f32(false, a1, false, b3, (short)0, acc[1][3], false, false);
        }
    }

    float* Outf = ((which == 0) ? P : Q) + (size_t)f * ROWS * ROWS;
    const int n  = lane & 15;
    const int mh = (lane >> 4) * 8;          // C/D: VGPR r -> rows r, r+8
    #pragma unroll
    for (int m = 0; m < 2; ++m) {
        #pragma unroll
        for (int t = 0; t < 4; ++t) {
            float* O = Outf + (size_t)(tmg * 32 + m * 16 + mh) * ROWS
                            + tng * 64 + t * 16 + n;
            #pragma unroll
            for (int r = 0; r < 8; ++r)
                O[(size_t)r * ROWS] = acc[m][t][r];
        }
    }
}

// ---------------------------------------------------------------------------
// Kernel 3: gather negatives from P/Q + norms, produce masked sims.
// One thread per (f,i,j).
// ---------------------------------------------------------------------------
__global__ void sims_kernel(const float* __restrict__ P,
                            const float* __restrict__ Q,
                            const float* __restrict__ Ni,
                            const float* __restrict__ Nj,
                            const int*  __restrict__ idxJ,
                            const int*  __restrict__ idxK,
                            const long long* __restrict__ ids,
                            float* __restrict__ simJ,
                            float* __restrict__ simK) {
    int t = blockIdx.x * blockDim.x + threadIdx.x;
    if (t >= F_ * B_ * B_) return;
    int j = t & 63;
    int i = (t >> 6) & 63;
    int f = t >> 12;

    float na2 = 0.f;
    const float* nia = Ni + (f * B_ + i) * C_;
    #pragma unroll
    for (int c = 0; c < C_; ++c) na2 += nia[c];
    float na = fmaxf(sqrtf(na2), EPSV);

    bool masked = (i == j) || (ids[i] == ids[j]);

    const int* ij = idxJ + ((size_t)(f * B_ + i) * B_ + j) * C_;
    const int* ik = idxK + ((size_t)(f * B_ + i) * B_ + j) * C_;
    const float* Pf  = P  + (size_t)f * ROWS * ROWS;
    const float* Qf  = Q  + (size_t)f * ROWS * ROWS;
    const float* njr = Nj + (f * B_ + j) * C_;
    const float* nir = Ni + (f * B_ + j) * C_;

    float dotj = 0.f, nj2 = 0.f, dotk = 0.f, nk2 = 0.f;
    #pragma unroll
    for (int c = 0; c < C_; ++c) {
        int cj = ij[c];
        int ck = ik[c];
        dotj += Pf[(size_t)(i * 16 + c) * ROWS + j * 16 + cj];
        nj2  += njr[cj];
        dotk += Qf[(size_t)(i * 16 + c) * ROWS + j * 16 + ck];
        nk2  += nir[ck];
    }

    size_t o = (size_t)(f * B_ + i) * B_ + j;
    if (masked) {
        simJ[o] = NEG_INF_V;
        simK[o] = NEG_INF_V;
    } else {
        simJ[o] = dotj / (na * fmaxf(sqrtf(nj2), EPSV)) * TEMP_INV;
        simK[o] = dotk / (na * fmaxf(sqrtf(nk2), EPSV)) * TEMP_INV;
    }
}

// ---------------------------------------------------------------------------
// Kernel 4: pos_sim + logsumexp per (f,i), block-reduce to scalar loss.
// Single block of 192 threads (F_*B_).
// ---------------------------------------------------------------------------
__global__ void loss_kernel(const float* __restrict__ P,
                            const float* __restrict__ Ni,
                            const float* __restrict__ Nj,
                            const float* __restrict__ simJ,
                            const float* __restrict__ simK,
                            float* __restrict__ out) {
    __shared__ float red[F_ * B_];
    int t = threadIdx.x;                       // 0..191
    int i = t & 63, f = t >> 6;

    const float* Pf  = P  + (size_t)f * ROWS * ROWS;
    const float* nia = Ni + (f * B_ + i) * C_;
    const float* nja = Nj + (f * B_ + i) * C_;
    float dot = 0.f, na2 = 0.f, nb2 = 0.f;
    #pragma unroll
    for (int c = 0; c < C_; ++c) {
        dot += Pf[(size_t)(i * 16 + c) * ROWS + i * 16 + c];   // diagonal
        na2 += nia[c];
        nb2 += nja[c];
    }
    float pos = dot / (fmaxf(sqrtf(na2), EPSV) * fmaxf(sqrtf(nb2), EPSV)) * TEMP_INV;

    const float* sj = simJ + (size_t)(f * B_ + i) * B_;
    const float* sk = simK + (size_t)(f * B_ + i) * B_;
    float m = pos;
    for (int j = 0; j < B_; ++j) {
        m = fmaxf(m, sj[j]);
        m = fmaxf(m, sk[j]);
    }
    float s = expf(pos - m);
    for (int j = 0; j < B_; ++j)
        s += expf(sj[j] - m) + expf(sk[j] - m);

    red[t] = logf(s) + m - pos;
    __syncthreads();
    if (t == 0) {
        float total = 0.f;
        for (int k = 0; k < F_ * B_; ++k) total += red[k];
        out[0] = total / (2.0f * B_);
    }
}

// ---------------------------------------------------------------------------
extern "C" void kernel_launch(void* const* d_in, const int* in_sizes, int n_in,
                              void* d_out, int out_size, void* d_ws, size_t ws_size,
                              hipStream_t stream) {
    const float*     Xi   = (const float*)d_in[0];      // [F,B,C,D] f32
    const float*     Xj   = (const float*)d_in[1];      // [F,B,C,D] f32
    const long long* ids  = (const long long*)d_in[2];  // [B] i64
    const int*       idxJ = (const int*)d_in[3];        // [F,B,B,C] i32
    const int*       idxK = (const int*)d_in[4];        // [F,B,B,C] i32
    float* out = (float*)d_out;

    // Workspace layout (floats): P(3M) | Q(3M) | Ni(3072) | Nj(3072) |
    //                            simJ(12288) | simK(12288)  ~= 25.3 MB
    float* P    = (float*)d_ws;
    float* Q    = P  + (size_t)F_ * ROWS * ROWS;
    float* Ni   = Q  + (size_t)F_ * ROWS * ROWS;
    float* Nj   = Ni + F_ * ROWS;
    float* simJ = Nj + F_ * ROWS;
    float* simK = simJ + F_ * B_ * B_;

    // 1) row norms: 6144 waves -> 768 blocks x 256 threads
    row_norms_kernel<<<768, 256, 0, stream>>>(Xi, Xj, Ni, Nj);

    // 2) WMMA GEMMs: 3 f x 2 matrices x (32 M-groups x 16 N-groups) = 3072
    //    waves, 8 waves/block -> 384 blocks x 256 threads
    gemm_wmma_kernel<<<384, 256, 0, stream>>>(Xi, Xj, P, Q);

    // 3) gathered negative sims: 12288 threads
    sims_kernel<<<48, 256, 0, stream>>>(P, Q, Ni, Nj, idxJ, idxK, ids, simJ, simK);

    // 4) final loss scalar
    loss_kernel<<<1, F_ * B_, 0, stream>>>(P, Ni, Nj, simJ, simK, out);
}